// replaceable_model_38886633898057
// MI455X (gfx1250) — compile-verified
//
#include <hip/hip_runtime.h>
#include <math.h>

#define BB 16
#define TT 32
#define NN 1500
#define FF 56
#define HIDN 64
#define DD 8
#define STEPS 31
#define INITL 4
#define OUTT 27

typedef float v2f __attribute__((ext_vector_type(2)));
typedef float v8f __attribute__((ext_vector_type(8)));

__device__ __forceinline__ float sigmoidf_(float x) { return 1.0f / (1.0f + __expf(-x)); }

// ---------------------------------------------------------------------------
// Degree vectors: dfinv[t][n] = rsqrt(1 + rowsum(adj[t], n))
//                 dbinv[t][n] = rsqrt(1 + colsum(adj[t], n))
// ---------------------------------------------------------------------------
__global__ void degree_row_kernel(const float* __restrict__ adj, float* __restrict__ dfinv) {
    int t = blockIdx.x;
    int wave = threadIdx.x >> 5;
    int lane = threadIdx.x & 31;
    int row = blockIdx.y * 8 + wave;
    if (row >= NN) return;
    const float* a = adj + ((long)t * NN + row) * NN;
    float s = 0.0f;
    for (int m = lane; m < NN; m += 32) s += a[m];
    for (int off = 16; off > 0; off >>= 1) s += __shfl_down(s, off, 32);
    if (lane == 0) dfinv[t * NN + row] = rsqrtf(1.0f + s);
}

__global__ void degree_col_kernel(const float* __restrict__ adj, float* __restrict__ dbinv) {
    int t = blockIdx.x;
    int col = blockIdx.y * 256 + threadIdx.x;
    if (col >= NN) return;
    const float* a = adj + (long)t * NN * NN + col;
    float s = 0.0f;
    for (int m = 0; m < NN; ++m) { s += *a; a += NN; }
    dbinv[t * NN + col] = rsqrtf(1.0f + s);
}

__global__ void mask_clear_kernel(int* __restrict__ mask) {
    int i = blockIdx.x * 256 + threadIdx.x;
    if (i < NN) mask[i] = 0;
}
__global__ void mask_set_kernel(const int* __restrict__ cells, int ncell, int* __restrict__ mask) {
    int i = blockIdx.x * 256 + threadIdx.x;
    if (i < ncell) {
        int c = cells[i];
        if (c >= 0 && c < NN) mask[c] = 1;
    }
}

// ---------------------------------------------------------------------------
// lap_gemm: O[b,n,f] = sum_m lap[n,m] * V[b,m,f]
//   lap[n,m] = (adj[n,m] + (n==m)) * dinv[n] * dinv[m]   (TRANSPOSED: adj[m,n])
// One wave computes a 32(M) x 64(cols) tile via V_WMMA_F32_16X16X4_F32.
// Columns c in [0, BB*C): b = c/C, f = c%C.
// All strides compile-time: VSM = per-m stride of V, VSB = per-batch stride.
// ---------------------------------------------------------------------------
template <int TRANSPOSED, int C, int VSM, int VSB>
__global__ __launch_bounds__(32) void lap_gemm_kernel(
    const float* __restrict__ adjt, const float* __restrict__ dinv,
    const float* __restrict__ V, float* __restrict__ O)
{
    int lane = threadIdx.x;
    int lm = lane & 15, half = lane >> 4;
    int m0 = blockIdx.x * 32;
    int c0 = blockIdx.y * 64;

    int r0 = m0 + lm;
    int r1 = m0 + 16 + lm;
    float dr0 = (r0 < NN) ? dinv[r0] : 0.0f;   // OOB rows contribute 0
    float dr1 = (r1 < NN) ? dinv[r1] : 0.0f;

    // 4 named stepping pointers for the B-column sub-tiles (keeps them GLOBAL)
    int c_0 = c0 + 0 * 16 + lm, c_1 = c0 + 1 * 16 + lm;
    int c_2 = c0 + 2 * 16 + lm, c_3 = c0 + 3 * 16 + lm;
    const float* pc0 = V + (long)(c_0 / C) * VSB + (c_0 % C) + (long)(half * 2) * VSM;
    const float* pc1 = V + (long)(c_1 / C) * VSB + (c_1 % C) + (long)(half * 2) * VSM;
    const float* pc2 = V + (long)(c_2 / C) * VSB + (c_2 % C) + (long)(half * 2) * VSM;
    const float* pc3 = V + (long)(c_3 / C) * VSB + (c_3 % C) + (long)(half * 2) * VSM;

    // stepping pointers for the A operand (lap rows) and the dinv stream
    const float* pa0;
    const float* pa1;
    if (TRANSPOSED) {
        pa0 = adjt + (long)(half * 2) * NN + r0;
        pa1 = adjt + (long)(half * 2) * NN + r1;
    } else {
        pa0 = adjt + (long)r0 * NN + half * 2;
        pa1 = adjt + (long)r1 * NN + half * 2;
    }
    const float* pd = dinv + half * 2;

    v8f acc0[4] = {};
    v8f acc1[4] = {};

    int ka = half * 2;
#pragma unroll 2
    for (int k0 = 0; k0 < NN; k0 += 4) {
        float dk0 = pd[0];
        float dk1 = pd[1];

        float x00, x01, x10, x11;
        if (TRANSPOSED) {
            x00 = pa0[0];  x01 = pa0[NN];
            x10 = pa1[0];  x11 = pa1[NN];
            __builtin_prefetch(pa0 + 32L * NN, 0, 1);
        } else {
            x00 = pa0[0];  x01 = pa0[1];
            x10 = pa1[0];  x11 = pa1[1];
            __builtin_prefetch(pa0 + 32, 0, 1);
        }
        v2f a0, a1;
        a0.x = (x00 + (r0 == ka     ? 1.0f : 0.0f)) * dr0 * dk0;
        a0.y = (x01 + (r0 == ka + 1 ? 1.0f : 0.0f)) * dr0 * dk1;
        a1.x = (x10 + (r1 == ka     ? 1.0f : 0.0f)) * dr1 * dk0;
        a1.y = (x11 + (r1 == ka + 1 ? 1.0f : 0.0f)) * dr1 * dk1;

        v2f b0, b1, b2, b3;
        b0.x = pc0[0]; b0.y = pc0[VSM];
        b1.x = pc1[0]; b1.y = pc1[VSM];
        b2.x = pc2[0]; b2.y = pc2[VSM];
        b3.x = pc3[0]; b3.y = pc3[VSM];

        acc0[0] = __builtin_amdgcn_wmma_f32_16x16x4_f32(false, a0, false, b0, (short)0, acc0[0], false, false);
        acc1[0] = __builtin_amdgcn_wmma_f32_16x16x4_f32(false, a1, false, b0, (short)0, acc1[0], false, false);
        acc0[1] = __builtin_amdgcn_wmma_f32_16x16x4_f32(false, a0, false, b1, (short)0, acc0[1], false, false);
        acc1[1] = __builtin_amdgcn_wmma_f32_16x16x4_f32(false, a1, false, b1, (short)0, acc1[1], false, false);
        acc0[2] = __builtin_amdgcn_wmma_f32_16x16x4_f32(false, a0, false, b2, (short)0, acc0[2], false, false);
        acc1[2] = __builtin_amdgcn_wmma_f32_16x16x4_f32(false, a1, false, b2, (short)0, acc1[2], false, false);
        acc0[3] = __builtin_amdgcn_wmma_f32_16x16x4_f32(false, a0, false, b3, (short)0, acc0[3], false, false);
        acc1[3] = __builtin_amdgcn_wmma_f32_16x16x4_f32(false, a1, false, b3, (short)0, acc1[3], false, false);

        pc0 += 4 * VSM; pc1 += 4 * VSM; pc2 += 4 * VSM; pc3 += 4 * VSM;
        pa0 += TRANSPOSED ? 4 * NN : 4;
        pa1 += TRANSPOSED ? 4 * NN : 4;
        pd += 4;
        ka += 4;
    }

    // ---- epilogue: C/D layout -> row = m + j + half*8, col = c0 + s*16 + lm
    bool full = (m0 + 32 <= NN);         // uniform per block
#pragma unroll
    for (int s = 0; s < 4; ++s) {
        int c = c0 + s * 16 + lm;
        int b = c / C, f = c % C;
        float* p = O + (long)b * (NN * C) + f + (long)(m0 + half * 8) * C;
        if (full) {
#pragma unroll
            for (int j = 0; j < 8; ++j) {
                p[j * C] = acc0[s][j];
                p[(16 + j) * C] = acc1[s][j];
            }
        } else {
#pragma unroll
            for (int j = 0; j < 8; ++j) {
                int row0 = m0 + j + half * 8;
                if (row0 < NN) p[j * C] = acc0[s][j];
                if (row0 + 16 < NN) p[(16 + j) * C] = acc1[s][j];
            }
        }
    }
}

// ---------------------------------------------------------------------------
// dense_gemm: O[r, ocol+c] = act( sum_k X[r,k] * W[k,c] + bias[c] )
// rows are a multiple of 16 (B*N = 24000); cols a multiple of 16.
// One wave = 16x16 tile. ACT: 0 = none, 1 = relu.
// K, LDW, LDX, LDO compile-time -> fully unrolled, immediate-offset loads.
// ---------------------------------------------------------------------------
template <int ACT, int K, int LDW, int LDX, int LDO>
__global__ __launch_bounds__(32) void dense_gemm_kernel(
    const float* __restrict__ X,
    const float* __restrict__ W,
    const float* __restrict__ bias,
    float* __restrict__ O, int ocol)
{
    int lane = threadIdx.x;
    int lm = lane & 15, half = lane >> 4;
    int r0 = blockIdx.x * 16;
    int c0 = blockIdx.y * 16;
    int c = c0 + lm;
    int r = r0 + lm;

    const float* xp = X + (long)r * LDX + half * 2;
    const float* wp = W + (long)(half * 2) * LDW + c;

    v8f acc = {};
#pragma unroll
    for (int k0 = 0; k0 < K; k0 += 4) {
        v2f a, b;
        a.x = xp[k0];
        a.y = xp[k0 + 1];
        b.x = wp[k0 * LDW];
        b.y = wp[(k0 + 1) * LDW];
        acc = __builtin_amdgcn_wmma_f32_16x16x4_f32(false, a, false, b, (short)0, acc, false, false);
    }

    float bv = bias[c];
    float* op = O + (long)(r0 + half * 8) * LDO + ocol + c;
#pragma unroll
    for (int j = 0; j < 8; ++j) {
        float v = acc[j] + bv;
        if (ACT == 1) v = fmaxf(v, 0.0f);
        op[j * LDO] = v;
    }
}

// ---------------------------------------------------------------------------
// attention: per node, lane_emb (8) + dir_emb (24) -> tmp[row*32 + 0..31]
// ---------------------------------------------------------------------------
__global__ void attention_kernel(
    const float* __restrict__ xt,                    // input + s*N*F; b stride = T*N*F
    const float* __restrict__ hf, const float* __restrict__ hb,
    const float* __restrict__ Wla, const float* __restrict__ bla,
    const float* __restrict__ Wle, const float* __restrict__ ble,
    const float* __restrict__ Wda, const float* __restrict__ bda,
    const float* __restrict__ Wde, const float* __restrict__ bde,
    float* __restrict__ tmp)
{
    int row = blockIdx.x * 256 + threadIdx.x;
    if (row >= BB * NN) return;
    int b = row / NN, n = row % NN;
    const float* xp = xt + ((long)b * TT * NN + n) * FF;

    float hfv[8], hbv[8], xl[8], xd[24];
#pragma unroll
    for (int i = 0; i < 8; ++i) {
        hfv[i] = hf[(long)row * 8 + i];
        hbv[i] = hb[(long)row * 8 + i];
        xl[i] = xp[8 + i];
    }
#pragma unroll
    for (int i = 0; i < 24; ++i) xd[i] = xp[16 + i];

    for (int o = 0; o < 8; ++o) {
        float a = bla[o], e = ble[o];
#pragma unroll
        for (int i = 0; i < 8; ++i) {
            a += hfv[i] * Wla[i * 8 + o] + hbv[i] * Wla[(8 + i) * 8 + o];
            e += xl[i] * Wle[i * 8 + o];
        }
        tmp[(long)row * 32 + o] = sigmoidf_(a) * e;
    }
    for (int o = 0; o < 24; ++o) {
        float a = bda[o], e = bde[o];
#pragma unroll
        for (int i = 0; i < 8; ++i)
            a += hfv[i] * Wda[i * 24 + o] + hbv[i] * Wda[(8 + i) * 24 + o];
#pragma unroll
        for (int i = 0; i < 24; ++i) e += xd[i] * Wde[i * 24 + o];
        tmp[(long)row * 32 + 8 + o] = sigmoidf_(a) * e;
    }
}

// ---------------------------------------------------------------------------
// GRU gate fusion: h_new = (1-z)*tanh(gi_n + r*gh_n) + z*h_space
// ---------------------------------------------------------------------------
__global__ void gru_kernel(const float* __restrict__ gi, const float* __restrict__ gh,
                           const float* __restrict__ hsp, float* __restrict__ h)
{
    long idx = (long)blockIdx.x * 256 + threadIdx.x;
    if (idx >= (long)BB * NN * HIDN) return;
    long row = idx / HIDN;
    int f = (int)(idx % HIDN);
    const float* gip = gi + row * 192;
    const float* ghp = gh + row * 192;
    float r = sigmoidf_(gip[f] + ghp[f]);
    float z = sigmoidf_(gip[64 + f] + ghp[64 + f]);
    float nn = tanhf(gip[128 + f] + r * ghp[128 + f]);
    h[idx] = (1.0f - z) * nn + z * hsp[idx];
}

// ---------------------------------------------------------------------------
// Output head with input_cells mask.
// ---------------------------------------------------------------------------
__global__ void out_kernel(const float* __restrict__ h, const float* __restrict__ Wout,
                           const float* __restrict__ bout, const int* __restrict__ mask,
                           const float* __restrict__ input, int s, float* __restrict__ out)
{
    int idx = blockIdx.x * 256 + threadIdx.x;
    if (idx >= BB * NN * DD) return;
    int dind = idx % DD;
    int n = (idx / DD) % NN;
    int b = idx / (DD * NN);
    float v;
    if (mask[n]) {
        v = input[(((long)b * TT + (s + 1)) * NN + n) * FF + dind];
    } else {
        const float* hp = h + ((long)b * NN + n) * HIDN;
        v = bout[dind];
        for (int k = 0; k < HIDN; ++k) v += hp[k] * Wout[k * DD + dind];
    }
    out[(((long)b * OUTT + (s - INITL)) * NN + n) * DD + dind] = v;
}

// ---------------------------------------------------------------------------
extern "C" void kernel_launch(void* const* d_in, const int* in_sizes, int n_in,
                              void* d_out, int out_size, void* d_ws, size_t ws_size,
                              hipStream_t stream)
{
    const float* input  = (const float*)d_in[0];
    const float* adj    = (const float*)d_in[1];
    const float* W_init = (const float*)d_in[2];  const float* b_init = (const float*)d_in[3];
    const float* W_fwd  = (const float*)d_in[4];  const float* b_fwd  = (const float*)d_in[5];
    const float* W_bwd  = (const float*)d_in[6];  const float* b_bwd  = (const float*)d_in[7];
    const float* W_mrg  = (const float*)d_in[8];  const float* b_mrg  = (const float*)d_in[9];
    const float* W_la   = (const float*)d_in[10]; const float* b_la   = (const float*)d_in[11];
    const float* W_le   = (const float*)d_in[12]; const float* b_le   = (const float*)d_in[13];
    const float* W_da   = (const float*)d_in[14]; const float* b_da   = (const float*)d_in[15];
    const float* W_de   = (const float*)d_in[16]; const float* b_de   = (const float*)d_in[17];
    const float* W_ih   = (const float*)d_in[18]; const float* b_ih   = (const float*)d_in[19];
    const float* W_hh   = (const float*)d_in[20]; const float* b_hh   = (const float*)d_in[21];
    const float* W_out  = (const float*)d_in[22]; const float* b_out  = (const float*)d_in[23];
    const int*   cells  = (const int*)d_in[24];
    int ncell = in_sizes[24];
    float* out = (float*)d_out;

    // ---- workspace layout (floats) ----
    float* ws = (float*)d_ws;
    size_t off = 0;
    float* dfinv = ws + off; off += (size_t)TT * NN;
    float* dbinv = ws + off; off += (size_t)TT * NN;
    int*   mask  = (int*)(ws + off); off += NN;
    float* h     = ws + off; off += (size_t)BB * NN * HIDN;
    float* hf    = ws + off; off += (size_t)BB * NN * DD;
    float* hb    = ws + off; off += (size_t)BB * NN * DD;
    float* tmp   = ws + off; off += (size_t)BB * NN * 32;
    float* spf   = ws + off; off += (size_t)BB * NN * HIDN;   // also holds S0 (B*N*56)
    float* spb   = ws + off; off += (size_t)BB * NN * HIDN;
    float* fhbh  = ws + off; off += (size_t)BB * NN * 2 * HIDN;
    float* hsp   = ws + off; off += (size_t)BB * NN * HIDN;
    float* gi    = ws + off; off += (size_t)BB * NN * 192;
    float* gh    = ws + off; off += (size_t)BB * NN * 192;

    const int MT = (NN + 31) / 32;        // 47 row-tiles for lap gemms
    const int RT = (BB * NN) / 16;        // 1500 row-tiles for dense gemms

    // ---- degree vectors + node mask ----
    degree_row_kernel<<<dim3(TT, (NN + 7) / 8), 256, 0, stream>>>(adj, dfinv);
    degree_col_kernel<<<dim3(TT, (NN + 255) / 256), 256, 0, stream>>>(adj, dbinv);
    mask_clear_kernel<<<(NN + 255) / 256, 256, 0, stream>>>(mask);
    mask_set_kernel<<<(ncell + 255) / 256, 256, 0, stream>>>(cells, ncell, mask);

    // ---- hidden init: relu((lap_f[0] @ x0) @ W_init + b_init) ----
    lap_gemm_kernel<0, FF, FF, TT * NN * FF><<<dim3(MT, (BB * FF) / 64), 32, 0, stream>>>(
        adj, dfinv, input, spf);
    dense_gemm_kernel<1, FF, HIDN, FF, HIDN><<<dim3(RT, HIDN / 16), 32, 0, stream>>>(
        spf, W_init, b_init, h, 0);

    // ---- scan over 31 steps ----
    for (int s = 0; s < STEPS; ++s) {
        const float* adjt = adj + (size_t)s * NN * NN;
        const float* dft = dfinv + s * NN;
        const float* dbt = dbinv + s * NN;
        const float* xt = input + (size_t)s * NN * FF;   // + b*T*N*F + n*F + f

        // hf/hb = lap_{f,b} @ dyn
        lap_gemm_kernel<0, DD, FF, TT * NN * FF><<<dim3(MT, (BB * DD) / 64), 32, 0, stream>>>(
            adjt, dft, xt, hf);
        lap_gemm_kernel<1, DD, FF, TT * NN * FF><<<dim3(MT, (BB * DD) / 64), 32, 0, stream>>>(
            adjt, dbt, xt, hb);

        // lane_emb + dir_emb -> tmp (B*N x 32)
        attention_kernel<<<(BB * NN + 255) / 256, 256, 0, stream>>>(
            xt, hf, hb, W_la, b_la, W_le, b_le, W_da, b_da, W_de, b_de, tmp);

        // spre_{f,b} = lap_{f,b} @ h
        lap_gemm_kernel<0, HIDN, HIDN, NN * HIDN><<<dim3(MT, (BB * HIDN) / 64), 32, 0, stream>>>(
            adjt, dft, h, spf);
        lap_gemm_kernel<1, HIDN, HIDN, NN * HIDN><<<dim3(MT, (BB * HIDN) / 64), 32, 0, stream>>>(
            adjt, dbt, h, spb);

        // fh|bh (relu) -> fhbh ; h_space = fhbh @ W_merge + b
        dense_gemm_kernel<1, HIDN, HIDN, HIDN, 2 * HIDN><<<dim3(RT, HIDN / 16), 32, 0, stream>>>(
            spf, W_fwd, b_fwd, fhbh, 0);
        dense_gemm_kernel<1, HIDN, HIDN, HIDN, 2 * HIDN><<<dim3(RT, HIDN / 16), 32, 0, stream>>>(
            spb, W_bwd, b_bwd, fhbh, HIDN);
        dense_gemm_kernel<0, 2 * HIDN, HIDN, 2 * HIDN, HIDN><<<dim3(RT, HIDN / 16), 32, 0, stream>>>(
            fhbh, W_mrg, b_mrg, hsp, 0);

        // GRU input/hidden projections
        dense_gemm_kernel<0, 32, 192, 32, 192><<<dim3(RT, 192 / 16), 32, 0, stream>>>(
            tmp, W_ih, b_ih, gi, 0);
        dense_gemm_kernel<0, HIDN, 192, HIDN, 192><<<dim3(RT, 192 / 16), 32, 0, stream>>>(
            hsp, W_hh, b_hh, gh, 0);

        // gates -> new hidden (overwrites h)
        gru_kernel<<<(int)(((long)BB * NN * HIDN + 255) / 256), 256, 0, stream>>>(gi, gh, hsp, h);

        // output head for steps >= INIT_LEN
        if (s >= INITL) {
            out_kernel<<<(BB * NN * DD + 255) / 256, 256, 0, stream>>>(
                h, W_out, b_out, mask, input, s, out);
        }
    }
    (void)n_in; (void)out_size; (void)ws_size;
}